// myFC2_420906795416
// MI455X (gfx1250) — compile-verified
//
#include <hip/hip_runtime.h>

// Fused grouped-1x1-conv + softmax-attention-collapse for MI455X (gfx1250).
// One kernel, no y materialization in HBM. fp32 WMMA (V_WMMA_F32_16X16X4_F32)
// for the grouped GEMM; online softmax in phase 1; GEMM recompute (L2-hot) in
// phase 2 plus the strided 's' channel GEMM; epilogue out = s + sum_r x*kv.
//
// Round-1 change: stage the B operand (weight fragments) through an explicit
// register array so the 16 b64 loads of a tile issue as one clause with a
// single s_wait_loadcnt, followed by 16 back-to-back v_wmma (C-chain only),
// instead of load->wait->wmma serialization per K-step.

typedef float v2f __attribute__((ext_vector_type(2)));
typedef float v8f __attribute__((ext_vector_type(8)));

// A-operand for V_WMMA_F32_16X16X4_F32: lane&15 = M row, lane>>4 selects K pair
// {0,1} or {2,3}; each lane loads 2 consecutive K floats per WMMA (b64 load).
__device__ __forceinline__ void load_areg(v2f* areg, const float* __restrict__ aRow,
                                          int ga, int khalf) {
  const float* p = aRow + ga * 64 + 2 * khalf;
#pragma unroll
  for (int kk = 0; kk < 16; ++kk)
    areg[kk] = *reinterpret_cast<const v2f*>(p + 4 * kk);
}

// 16x16 output tile, K=64 via 16 chained f32 WMMAs; bias folded into C init.
// All 16 B fragments are preloaded so the WMMA chain runs without per-step
// memory waits.
__device__ __forceinline__ v8f tile_gemm(const v2f* areg, const float* __restrict__ Wp,
                                         float bb, int khalf) {
  v2f breg[16];
#pragma unroll
  for (int kk = 0; kk < 16; ++kk)
    breg[kk] = *reinterpret_cast<const v2f*>(Wp + 4 * kk + 2 * khalf);
  v8f acc = {bb, bb, bb, bb, bb, bb, bb, bb};
#pragma unroll
  for (int kk = 0; kk < 16; ++kk) {
    acc = __builtin_amdgcn_wmma_f32_16x16x4_f32(
        /*neg_a=*/false, areg[kk], /*neg_b=*/false, breg[kk],
        /*c_mod=*/(short)0, acc, /*reuse_a=*/false, /*reuse_b=*/false);
  }
  return acc;
}

// D layout: lane&15 = N (channel), VGPR i -> row M = i + 8*(lane>>4).
__device__ __forceinline__ void store_tile(float* buf, int nt, int lane, v8f acc) {
  const int n = lane & 15, kh = lane >> 4;
#pragma unroll
  for (int i = 0; i < 8; ++i)
    buf[(i + 8 * kh) * 128 + nt * 16 + n] = acc[i];
}

__global__ __launch_bounds__(256) void fc2_fused(const float* __restrict__ in,   // (T,B,4096)
                                                 const float* __restrict__ W,    // (16384,64)
                                                 const float* __restrict__ bias, // (16384)
                                                 float* __restrict__ out) {      // (T,B,4096)
  __shared__ float ytile[4][16 * 128]; // x[n,o',r] for current 128-wide o block
  __shared__ float stile[16 * 128];    // s[n,o'] for current block (phase 2)
  __shared__ float xsumb[16 * 128];    // sum over reps (phase 1)
  __shared__ float kvbuf[64];          // kv[row][r]

  const int tid = threadIdx.x;
  const int lane = tid & 31;
  const int wave = tid >> 5;
  const int n0 = blockIdx.x * 16;   // 16 rows (b,t) per workgroup
  const int bIdx = n0 >> 10;        // n = b*1024 + t
  const int t0 = n0 & 1023;
  const int khalf = lane >> 4;
  // per-lane input row pointer: input[(t0+row)*4 + b][*]
  const float* aRow = in + ((size_t)((t0 + (lane & 15)) * 4 + bIdx)) * 4096;

  const int rr = wave & 3;      // rep this wave computes x-tiles for
  const int ntBase = wave >> 2; // nt parity (two waves cover 8 n-tiles per rep)

  // online softmax state for (row = tid>>2, r = tid&3), tid<64
  float m = -3.0e38f, dsum = 0.0f, num = 0.0f;

  // -------- Phase 1: streaming stats (online softmax + kv numerator) --------
  for (int jj = 0; jj < 32; ++jj) {
    const int o_base = jj * 128;
    {
      const int ga = rr * 16 + (jj >> 1); // input group for (rep rr, o block)
      v2f areg[16];
      load_areg(areg, aRow, ga, khalf);
#pragma unroll
      for (int k4 = 0; k4 < 4; ++k4) {
        const int nt = ntBase + 2 * k4;
        const int ch = rr * 4096 + o_base + nt * 16 + (lane & 15);
        v8f acc = tile_gemm(areg, W + (size_t)ch * 64, bias[ch], khalf);
        store_tile(ytile[rr], nt, lane, acc);
      }
    }
    __syncthreads();
#pragma unroll
    for (int i = 0; i < 8; ++i) { // xsum over reps, all 256 threads
      const int e = i * 256 + tid;
      xsumb[e] = ytile[0][e] + ytile[1][e] + ytile[2][e] + ytile[3][e];
    }
    __syncthreads();
    if (tid < 64) { // one thread per (row, rep): online softmax update
      const int prow = tid >> 2, pr = tid & 3;
      const float* yv = &ytile[pr][prow * 128];
      const float* xs = &xsumb[prow * 128];
      for (int o = 0; o < 128; ++o) {
        const float v = yv[o], x = xs[o];
        if (v > m) { const float sc = __expf(m - v); dsum *= sc; num *= sc; m = v; }
        const float e = __expf(v - m);
        dsum += e;
        num += e * x;
      }
    }
    __syncthreads();
  }
  if (tid < 64) kvbuf[tid] = num / dsum; // kv[row*4 + r]
  __syncthreads();

  // -------- Phase 2: recompute (L2-hot) + s-GEMM + final output --------
  for (int jj = 0; jj < 32; ++jj) {
    const int o_base = jj * 128;
    {
      const int ga = rr * 16 + (jj >> 1);
      v2f areg[16];
      load_areg(areg, aRow, ga, khalf);
#pragma unroll
      for (int k4 = 0; k4 < 4; ++k4) {
        const int nt = ntBase + 2 * k4;
        const int ch = rr * 4096 + o_base + nt * 16 + (lane & 15);
        v8f acc = tile_gemm(areg, W + (size_t)ch * 64, bias[ch], khalf);
        store_tile(ytile[rr], nt, lane, acc);
      }
      // s tile: s[n,o] = y[ch=4o]; gathered-weight GEMM, group = (o/64)
      const int gs = 2 * jj + (wave >> 2);
      v2f sreg[16];
      load_areg(sreg, aRow, gs, khalf);
      const int chs = 4 * (o_base + wave * 16 + (lane & 15));
      v8f accs = tile_gemm(sreg, W + (size_t)chs * 64, bias[chs], khalf);
      store_tile(stile, wave, lane, accs);
    }
    __syncthreads();
#pragma unroll
    for (int i = 0; i < 8; ++i) { // out = s + sum_r x*kv ; coalesced stores
      const int e = i * 256 + tid;
      const int prow = e >> 7;
      const int oo = e & 127;
      const float* kvp = &kvbuf[prow * 4];
      const float val = stile[e] + ytile[0][e] * kvp[0] + ytile[1][e] * kvp[1] +
                        ytile[2][e] * kvp[2] + ytile[3][e] * kvp[3];
      out[((size_t)((t0 + prow) * 4 + bIdx)) * 4096 + o_base + oo] = val;
    }
    __syncthreads();
  }
}

extern "C" void kernel_launch(void* const* d_in, const int* in_sizes, int n_in,
                              void* d_out, int out_size, void* d_ws, size_t ws_size,
                              hipStream_t stream) {
  (void)in_sizes; (void)n_in; (void)d_ws; (void)ws_size; (void)out_size;
  const float* in = (const float*)d_in[0];   // (1024, 4, 4096) f32
  const float* W = (const float*)d_in[1];    // (16384, 64) f32
  const float* bias = (const float*)d_in[2]; // (16384,) f32
  float* out = (float*)d_out;                // (1024, 4, 4096) f32
  // 4096 rows / 16 rows per workgroup = 256 workgroups, 256 threads (8 waves)
  fc2_fused<<<256, 256, 0, stream>>>(in, W, bias, out);
}